// RBF_61151744361061
// MI455X (gfx1250) — compile-verified
//
#include <hip/hip_runtime.h>

typedef __attribute__((ext_vector_type(2))) float v2f;
typedef __attribute__((ext_vector_type(8))) float v8f;

#define RBF_BETA 0.04f
#define LOG2E    1.44269504088896340736f
#define NT       4   // N-tiles (of 16 centres) per wave

// out[b][n][0] = exp(-|x[b]-c[n]|^2/beta), out[b][n][1] = same for x_delayed.
// Rank-4 bilinear expansion:
//   -|x-c|^2/beta * log2e = A(x)·B(c),
//   A = [x0, x1, x0^2+x1^2, 1],  B = [k2*c0, k2*c1, k1, k1*(c0^2+c1^2)]
// computed with V_WMMA_F32_16X16X4_F32; epilogue is one v_exp_f32 per element.
// NSZ_C != 0 specializes the row stride so all loads/stores in the unrolled
// tile loop use one base address + compile-time immediate offsets.
template <int NSZ_C>
__global__ __launch_bounds__(256) void rbf_wmma_kernel(
    const float* __restrict__ x,        // [B,2]
    const float* __restrict__ xd,       // [B,2]
    const float* __restrict__ centres,  // [N,2]
    float* __restrict__ out,            // [B,N,2]
    int Bsz, int NszRT)
{
    const int Nsz = NSZ_C ? NSZ_C : NszRT;

    const int lane   = threadIdx.x & 31;
    const int waveIn = threadIdx.x >> 5;
    const int waveG  = blockIdx.x * 8 + waveIn;

    const int nSuper = Nsz / (16 * NT);     // number of N-supertiles
    const int mTile  = waveG / nSuper;      // which 16-row batch tile
    const int nSupI  = waveG % nSuper;      // which group of NT n-tiles
    if (mTile * 16 >= Bsz) return;

    const int  col = lane & 15;             // M index (A) / N index (B, D)
    const bool hi  = lane >= 16;

    // ---- A operands: row m = [x0, x1, x0^2+x1^2, 1] ----
    const int m  = mTile * 16 + col;
    const float x0 = x[2 * m],  x1 = x[2 * m + 1];
    const float y0 = xd[2 * m], y1 = xd[2 * m + 1];
    v2f aCur, aDel;
    if (!hi) {                              // lanes 0-15: K = {0,1}
        aCur = v2f{x0, x1};
        aDel = v2f{y0, y1};
    } else {                                // lanes 16-31: K = {2,3}
        aCur = v2f{x0 * x0 + x1 * x1, 1.0f};
        aDel = v2f{y0 * y0 + y1 * y1, 1.0f};
    }

    const float k2 = 2.0f * LOG2E / RBF_BETA;   //  72.1347...
    const float k1 = -LOG2E / RBF_BETA;         // -36.0673...

    const int rowBase = mTile * 16 + (hi ? 8 : 0);
    const int nBase   = nSupI * (NT * 16);

    // One base pointer each; all subsequent accesses are immediate offsets
    // when NSZ_C is a compile-time constant.
    const float* __restrict__ cptr = centres + 2 * (size_t)(nBase + col);
    float*       __restrict__ optr = out + 2 * ((size_t)rowBase * (size_t)Nsz
                                                + (size_t)(nBase + col));

    #pragma unroll
    for (int t = 0; t < NT; ++t) {
        // ---- B operand: col n = [k2*c0, k2*c1, k1, k1*(c0^2+c1^2)] ----
        const float c0 = cptr[t * 32 + 0];      // +t*128 bytes (immediate)
        const float c1 = cptr[t * 32 + 1];
        v2f b;
        if (!hi) b = v2f{c0 * k2, c1 * k2};             // K = {0,1}
        else     b = v2f{k1, (c0 * c0 + c1 * c1) * k1}; // K = {2,3}

        v8f accC = {};
        v8f accD = {};
        accC = __builtin_amdgcn_wmma_f32_16x16x4_f32(
            false, aCur, false, b, (short)0, accC, false, false);
        accD = __builtin_amdgcn_wmma_f32_16x16x4_f32(
            false, aDel, false, b, (short)0, accD, false, false);

        // ---- epilogue: exp2 + b64 stores at immediate offsets ----
        #pragma unroll
        for (int r = 0; r < 8; ++r) {
            const float o0 = __builtin_amdgcn_exp2f(accC[r]);
            const float o1 = __builtin_amdgcn_exp2f(accD[r]);
            float2 v; v.x = o0; v.y = o1;
            // byte offset = r*Nsz*8 + t*128 : compile-time when NSZ_C != 0
            *reinterpret_cast<float2*>(optr + (size_t)r * (size_t)Nsz * 2
                                            + (size_t)t * 32) = v;
        }
    }
}

extern "C" void kernel_launch(void* const* d_in, const int* in_sizes, int n_in,
                              void* d_out, int out_size, void* d_ws, size_t ws_size,
                              hipStream_t stream)
{
    const float* x       = (const float*)d_in[0];  // [B,2]
    const float* xd      = (const float*)d_in[1];  // [B,2]
    const float* centres = (const float*)d_in[2];  // [N,2]
    float* out           = (float*)d_out;          // [B,N,2]

    const int Bsz = in_sizes[0] / 2;   // 2048
    const int Nsz = in_sizes[2] / 2;   // 8192

    const int mTiles = Bsz / 16;               // 128
    const int nSuper = Nsz / (16 * NT);        // 128
    const int totalWaves = mTiles * nSuper;    // 16384
    const int blocks = (totalWaves + 7) / 8;   // 8 waves (256 thr) per block

    if (Nsz == 8192) {
        rbf_wmma_kernel<8192><<<blocks, 256, 0, stream>>>(x, xd, centres, out,
                                                          Bsz, Nsz);
    } else {
        rbf_wmma_kernel<0><<<blocks, 256, 0, stream>>>(x, xd, centres, out,
                                                       Bsz, Nsz);
    }
}